// StateSpaceModelLayer_15831249453528
// MI455X (gfx1250) — compile-verified
//
#include <hip/hip_runtime.h>

#define L_SEQ 32768
#define GOFF  (L_SEQ + 128)    // KR reversal pivot: KR[x] = K[GOFF - x]
#define PADN  (L_SEQ + 1024)   // KR and U padded buffer sizes (floats)
#define NSEG  4                // split-m0 segments for the conv reduction

typedef float v2f __attribute__((ext_vector_type(2)));
typedef float v8f __attribute__((ext_vector_type(8)));

__device__ __forceinline__ v8f wmma4(v2f a, v2f b, v8f c) {
  // D = A(16x4,f32) * B(4x16,f32) + C(16x16,f32)
  return __builtin_amdgcn_wmma_f32_16x16x4_f32(false, a, false, b, (short)0, c,
                                               false, false);
}

// ---------------------------------------------------------------------------
// Kernel 1: bilinear discretization via Gauss-Jordan on [I-hA | I+hA | step*B]
// One workgroup, 1024 threads. P = I-hA is strongly diagonally dominant
// (h <= 0.05, |A_offdiag| ~ N(0,1/128)), so no pivoting is required.
// ---------------------------------------------------------------------------
__global__ __launch_bounds__(1024)
void k_disc(const float* __restrict__ A, const float* __restrict__ B,
            const float* __restrict__ log_step,
            float* __restrict__ ab, float* __restrict__ bb) {
  __shared__ float Aug[128][257];
  const int tid = threadIdx.x;
  const float step = __expf(log_step[0]);
  const float h = 0.5f * step;

  for (int idx = tid; idx < 128 * 128; idx += 1024) {
    int r = idx >> 7, c = idx & 127;
    float a = A[idx];
    float d = (r == c) ? 1.0f : 0.0f;
    Aug[r][c]       = d - h * a;   // P = I - (step/2) A
    Aug[r][128 + c] = d + h * a;   // Q = I + (step/2) A
  }
  for (int r = tid; r < 128; r += 1024) Aug[r][256] = step * B[r];
  __syncthreads();

  const int lane8 = tid & 7;   // 8 column-threads per row
  const int row   = tid >> 3;  // 0..127
  for (int k = 0; k < 128; ++k) {
    float inv = 1.0f / Aug[k][k];
    if (row == k) {
      for (int c = k + 1 + lane8; c < 257; c += 8) Aug[k][c] *= inv;
    }
    __syncthreads();
    if (row != k) {
      float f = Aug[row][k];
      for (int c = k + 1 + lane8; c < 257; c += 8)
        Aug[row][c] -= f * Aug[k][c];
    }
    __syncthreads();
  }
  for (int idx = tid; idx < 128 * 128; idx += 1024) {
    int r = idx >> 7, c = idx & 127;
    ab[idx] = Aug[r][128 + c];          // ab = P^-1 Q
  }
  for (int r = tid; r < 128; r += 1024) bb[r] = Aug[r][256];  // bb = P^-1 step B
}

// ---------------------------------------------------------------------------
// Kernel 2: Block0[:,t] = ab^t * bb for t=0..127 (sequential matvec scan).
// One workgroup, 512 threads; ab cached in LDS; 4-lane split dot + shuffles.
// Block0 stored row-major [state k][col t].
// ---------------------------------------------------------------------------
__global__ __launch_bounds__(512)
void k_block0(const float* __restrict__ ab, const float* __restrict__ bb,
              float* __restrict__ B0) {
  __shared__ float As[128 * 128];
  __shared__ float xbuf[2][128];
  const int tid = threadIdx.x;
  for (int i = tid; i < 128 * 128; i += 512) As[i] = ab[i];
  if (tid < 128) xbuf[0][tid] = bb[tid];
  __syncthreads();

  const int r = tid >> 2, q = tid & 3;
  int cur = 0;
  for (int t = 0; t < 128; ++t) {
    if (tid < 128) B0[tid * 128 + t] = xbuf[cur][tid];
    float p = 0.0f;
    const float* rowp = As + r * 128 + q * 32;
    const float* xc   = xbuf[cur] + q * 32;
#pragma unroll
    for (int c = 0; c < 32; ++c) p += rowp[c] * xc[c];
    p += __shfl_xor(p, 1);
    p += __shfl_xor(p, 2);
    xbuf[cur ^ 1][r] = p;
    __syncthreads();
    cur ^= 1;
  }
}

// ---------------------------------------------------------------------------
// Kernel 3: generic fp32 GEMM  Cm(Mrows x 128) = Am(Mrows x 128) * Bm(128x128)
// via V_WMMA_F32_16X16X4_F32. One 16x16 tile per wave; blockDim = 256.
// Used for the 7 squarings (ab -> ab^128) and for K = R @ Block0.
// ---------------------------------------------------------------------------
__global__ __launch_bounds__(256)
void k_mm(const float* __restrict__ Am, const float* __restrict__ Bm,
          float* __restrict__ Cm, int Mrows) {
  const int W  = blockIdx.x * 8 + (threadIdx.x >> 5);
  const int ti = W >> 3;          // row tile
  const int tj = W & 7;           // col tile (128/16 = 8)
  if (ti * 16 >= Mrows) return;
  const int l  = threadIdx.x & 31;
  const int lm = l & 15, lh = l >> 4;

  v8f acc = {};
  const float* arow = Am + (ti * 16 + lm) * 128 + 2 * lh;  // A[M=lm][K=..]
  const float* bcol = Bm + (2 * lh) * 128 + tj * 16 + lm;  // B[K=..][N=lm]
#pragma unroll 8
  for (int c4 = 0; c4 < 32; ++c4) {
    v2f a = *(const v2f*)(arow + 4 * c4);      // K = 4c4 + 2lh + {0,1}, aligned
    v2f b;
    b.x = bcol[(4 * c4 + 0) * 128];
    b.y = bcol[(4 * c4 + 1) * 128];
    acc = wmma4(a, b, acc);
  }
#pragma unroll
  for (int j = 0; j < 8; ++j) {
    int rowo = ti * 16 + j + 8 * lh;           // C/D layout: M = j + 8*(l/16)
    Cm[rowo * 128 + tj * 16 + lm] = acc[j];
  }
}

// ---------------------------------------------------------------------------
// Kernel 4: row scan  R[j] = C * M^j  (M = ab^128), j = 0..255.
// g_{j+1}[c] = sum_k g_j[k] * M[k][c]; M transposed into LDS once.
// ---------------------------------------------------------------------------
__global__ __launch_bounds__(512)
void k_gscan(const float* __restrict__ Mm, const float* __restrict__ Cv,
             float* __restrict__ R) {
  __shared__ float Mt[128 * 128];  // Mt[c][k] = M[k][c]
  __shared__ float gbuf[2][128];
  const int tid = threadIdx.x;
  for (int i = tid; i < 128 * 128; i += 512) {
    int k = i >> 7, c = i & 127;
    Mt[c * 128 + k] = Mm[i];
  }
  if (tid < 128) gbuf[0][tid] = Cv[tid];
  __syncthreads();

  const int c = tid >> 2, q = tid & 3;
  int cur = 0;
  for (int j = 0; j < 256; ++j) {
    if (tid < 128) R[j * 128 + tid] = gbuf[cur][tid];
    float p = 0.0f;
    const float* mc = Mt + c * 128 + q * 32;
    const float* gp = gbuf[cur] + q * 32;
#pragma unroll
    for (int k = 0; k < 32; ++k) p += gp[k] * mc[k];
    p += __shfl_xor(p, 1);
    p += __shfl_xor(p, 2);
    gbuf[cur ^ 1][c] = p;
    __syncthreads();
    cur ^= 1;
  }
}

// ---------------------------------------------------------------------------
// Kernel 5: build zero-padded buffers.
//   KR[x] = K[GOFF - x] (0 outside [0,L))   -- reversed kernel for Toeplitz B
//   U[x]  = u[x - 128]  (0 outside [0,L))   -- left/right padded input
// ---------------------------------------------------------------------------
__global__ void k_pad(const float* __restrict__ K, const float* __restrict__ u,
                      float* __restrict__ KR, float* __restrict__ U) {
  int idx = blockIdx.x * blockDim.x + threadIdx.x;
  if (idx >= PADN) return;
  int s = GOFF - idx;
  KR[idx] = (s >= 0 && s < L_SEQ) ? K[s] : 0.0f;
  int m = idx - 128;
  U[idx] = (m >= 0 && m < L_SEQ) ? u[m] : 0.0f;
}

// ---------------------------------------------------------------------------
// Kernel 6: causal convolution as block-Toeplitz WMMA GEMM, split 4 ways
// along the reduction (m0) axis. Each wave handles one (tile T, segment s):
// tile T covers 512 outputs y[t0 .. t0+511], t0 = 512*T, as TWO 16x16 WMMA
// tiles (i-halves) sharing one B fragment -> 2 independent accumulator
// chains (ILP) and half the KR traffic.
//   Y[i][n] = y[t0 + 32n + i] = sum_{m0,k} U[128+m0+i+k] * KR[cb + k - 32n]
// with cb = GOFF - t0 + m0; chunks m0 = -128 + 128c, c in [c0,c1) of the
// tile's chunk count. Partials go to P[s][tau]; k_reduce sums in fixed order
// (deterministic, no float atomics).
// ---------------------------------------------------------------------------
__global__ __launch_bounds__(256)
void k_conv(const float* __restrict__ U, const float* __restrict__ KR,
            float* __restrict__ P) {
  const int W = blockIdx.x * 8 + (threadIdx.x >> 5);  // 0..255
  const int T = W >> 2;                               // tile 0..63
  const int s = W & (NSEG - 1);                       // segment 0..3
  const int t0 = T * 512;
  const int l  = threadIdx.x & 31;
  const int lm = l & 15, lh = l >> 4;

  const int m0cap = t0 + 480;
  const int m0end = (m0cap < L_SEQ - 128) ? m0cap : (L_SEQ - 128);
  const int nc = (m0end + 256) >> 7;        // number of 128-wide chunks
  const int c0 = (s * nc) >> 2;
  const int c1 = ((s + 1) * nc) >> 2;

  v8f acc0 = {}, acc1 = {};
  for (int c = c0; c < c1; ++c) {
    const int m0 = -128 + (c << 7);
    const float* ub = U + 128 + m0 + lm + 2 * lh;              // A[i][k]
    const float* kb = KR + (GOFF - t0 + m0) + 2 * lh - 32 * lm;  // B[k][n]
#pragma unroll 8
    for (int c4 = 0; c4 < 32; ++c4) {
      v2f b = *(const v2f*)(kb + 4 * c4);   // even index: aligned b64
      v2f a0, a1;                           // u parity varies: b32 loads
      a0.x = ub[4 * c4 + 0];
      a0.y = ub[4 * c4 + 1];
      a1.x = ub[4 * c4 + 16];
      a1.y = ub[4 * c4 + 17];
      acc0 = wmma4(a0, b, acc0);            // rows i = 0..15
      acc1 = wmma4(a1, b, acc1);            // rows i = 16..31
    }
  }
  // tau = t0 + 32*n + 16*h + (j + 8*lh), n = lm; contiguous in j
  float* Pb = P + s * L_SEQ + t0 + 32 * lm + 8 * lh;
#pragma unroll
  for (int j = 0; j < 8; ++j) Pb[j] = acc0[j];
#pragma unroll
  for (int j = 0; j < 8; ++j) Pb[16 + j] = acc1[j];
}

// ---------------------------------------------------------------------------
// Kernel 7: deterministic reduction of the NSEG partials + D*u skip term.
// ---------------------------------------------------------------------------
__global__ void k_reduce(const float* __restrict__ P, const float* __restrict__ u,
                         const float* __restrict__ Dp, float* __restrict__ y) {
  int t = blockIdx.x * blockDim.x + threadIdx.x;
  if (t >= L_SEQ) return;
  float v = P[t] + P[L_SEQ + t] + P[2 * L_SEQ + t] + P[3 * L_SEQ + t];
  y[t] = v + Dp[0] * u[t];
}

// ---------------------------------------------------------------------------
extern "C" void kernel_launch(void* const* d_in, const int* in_sizes, int n_in,
                              void* d_out, int out_size, void* d_ws,
                              size_t ws_size, hipStream_t stream) {
  (void)in_sizes; (void)n_in; (void)out_size; (void)ws_size;
  const float* u  = (const float*)d_in[0];
  const float* A  = (const float*)d_in[1];
  const float* B  = (const float*)d_in[2];
  const float* C  = (const float*)d_in[3];
  const float* D  = (const float*)d_in[4];
  const float* ls = (const float*)d_in[5];
  float* y = (float*)d_out;

  float* ws = (float*)d_ws;
  float* ab = ws;                    // 16384
  float* bb = ab + 16384;            //   128
  float* B0 = bb + 128;              // 16384
  float* W0 = B0 + 16384;            // 16384
  float* W1 = W0 + 16384;            // 16384
  float* R  = W1 + 16384;            // 32768
  float* K  = R  + 32768;            // 32768
  float* KR = K  + 32768;            // PADN
  float* U  = KR + PADN;             // PADN
  float* P  = U  + PADN;             // NSEG * L_SEQ  (total ~1.3 MB)

  // 1) discretize (ab, bb)
  k_disc<<<1, 1024, 0, stream>>>(A, B, ls, ab, bb);
  // 2) Block0 = [bb, ab*bb, ..., ab^127*bb]
  k_block0<<<1, 512, 0, stream>>>(ab, bb, B0);
  // 3) M = ab^128 by 7 WMMA squarings (ping-pong W0/W1; result in W0)
  k_mm<<<8, 256, 0, stream>>>(ab, ab, W0, 128);   // ab^2
  k_mm<<<8, 256, 0, stream>>>(W0, W0, W1, 128);   // ab^4
  k_mm<<<8, 256, 0, stream>>>(W1, W1, W0, 128);   // ab^8
  k_mm<<<8, 256, 0, stream>>>(W0, W0, W1, 128);   // ab^16
  k_mm<<<8, 256, 0, stream>>>(W1, W1, W0, 128);   // ab^32
  k_mm<<<8, 256, 0, stream>>>(W0, W0, W1, 128);   // ab^64
  k_mm<<<8, 256, 0, stream>>>(W1, W1, W0, 128);   // ab^128
  // 4) R[j] = C * (ab^128)^j, j=0..255
  k_gscan<<<1, 512, 0, stream>>>(W0, C, R);
  // 5) K (256x128 row-major == flat K[0..32767]) = R @ Block0  (WMMA GEMM)
  k_mm<<<16, 256, 0, stream>>>(R, B0, K, 256);
  // 6) padded/reversed buffers for the Toeplitz convolution
  k_pad<<<(PADN + 255) / 256, 256, 0, stream>>>(K, u, KR, U);
  // 7) partial causal conv (WMMA block-Toeplitz GEMM, 4-way split-m0)
  k_conv<<<32, 256, 0, stream>>>(U, KR, P);
  // 8) deterministic reduce + D*u
  k_reduce<<<(L_SEQ + 255) / 256, 256, 0, stream>>>(P, u, D, y);
}